// srm_38457137168911
// MI455X (gfx1250) — compile-verified
//
#include <hip/hip_runtime.h>
#include <math.h>

// Problem constants (from reference)
#define BB    64
#define QLEN  32
#define DLEN  1024
#define EDIM  300
#define NBINS 11
#define EP    304                    // padded LDS row stride (floats); 1216B = 16B aligned
#define NTOK  (BB*QLEN + BB*DLEN)    // 67584 tokens needing norms
#define DTILE 64                     // doc rows per tile
#define CHK_B 75                     // 16B chunks per row (300 floats)

typedef float v2f __attribute__((ext_vector_type(2)));
typedef float v8f __attribute__((ext_vector_type(8)));
typedef int   v4i __attribute__((vector_size(16)));   // matches builtin pointee type

// ---------------------------------------------------------------------------
// CDNA5 async global->LDS copy (16B per lane), with sync fallback
// ---------------------------------------------------------------------------
#if defined(__has_builtin)
#if __has_builtin(__builtin_amdgcn_global_load_async_to_lds_b128)
#define HAVE_ASYNC_LDS 1
#endif
#endif

static __device__ __forceinline__ void cp16_to_lds(const float* g, float* l)
{
#ifdef HAVE_ASYNC_LDS
    __builtin_amdgcn_global_load_async_to_lds_b128(
        (__attribute__((address_space(1))) v4i*)(uintptr_t)g,
        (__attribute__((address_space(3))) v4i*)(unsigned int)(uintptr_t)l,
        0, 0);
#else
    *(float4*)l = *(const float4*)g;
#endif
}

static __device__ __forceinline__ void async_fence_and_barrier()
{
#ifdef HAVE_ASYNC_LDS
#if __has_builtin(__builtin_amdgcn_s_wait_asynccnt)
    __builtin_amdgcn_s_wait_asynccnt(0);
#else
    asm volatile("s_wait_asynccnt 0x0" ::: "memory");
#endif
#endif
    __syncthreads();
}

// ---------------------------------------------------------------------------
// Kernel 1: inverse L2 norm per gathered token row (one wave32 per token)
// ---------------------------------------------------------------------------
__global__ __launch_bounds__(256) void token_inv_norms(
    const int* __restrict__ iq, const int* __restrict__ idd,
    const float* __restrict__ emb, float* __restrict__ inv)
{
    const int wave = blockIdx.x * 8 + (threadIdx.x >> 5);
    const int lane = threadIdx.x & 31;
    int tok;
    if (wave < BB*QLEN) tok = iq[wave];
    else                tok = idd[wave - BB*QLEN];
    const float* row = emb + (size_t)tok * EDIM;
    float s = 0.f;
    for (int i = lane; i < EDIM; i += 32) { float v = row[i]; s += v * v; }
#pragma unroll
    for (int off = 16; off >= 1; off >>= 1) s += __shfl_xor(s, off);
    if (lane == 0) inv[wave] = 1.0f / fmaxf(sqrtf(s), 1e-12f);
}

// ---------------------------------------------------------------------------
// Kernel 2: fused KNRM forward, one workgroup (8 waves) per batch element.
//   f32 WMMA for sim, double-buffered async-to-LDS doc-tile staging,
//   normalization folded into the WMMA output (raw rows staged).
// ---------------------------------------------------------------------------
__global__ __launch_bounds__(256) void knrm_forward(
    const int* __restrict__ iq, const int* __restrict__ idd,
    const float* __restrict__ mask_q, const float* __restrict__ mask_d,
    const float* __restrict__ emb, const float* __restrict__ attn_w,
    const float* __restrict__ idf_w, const float* __restrict__ idf_b,
    const float* __restrict__ alpha_p, const float* __restrict__ mu_p,
    const float* __restrict__ sigma_p, const float* __restrict__ dense_w,
    const float* __restrict__ dense_b,
    const float* __restrict__ invq, const float* __restrict__ invd,
    float* __restrict__ out)
{
    extern __shared__ float sm[];
    float* As   = sm;                        // 32  x EP (raw q rows)
    float* Bs   = As + QLEN * EP;            // 2 x 64 x EP (raw d rows, 2 buffers)
    float* accN = Bs + 2 * DTILE * EP;       // [4][32][11]
    float* accD = accN + 4 * QLEN * NBINS;   // [4][32][11]
    float* red  = accD + 4 * QLEN * NBINS;   // 352 log-pool + 11 feats

    const int b    = blockIdx.x;
    const int tid  = threadIdx.x;
    const int wave = tid >> 5;
    const int lane = tid & 31;
    const int half = lane >> 4;
    const int l16  = lane & 15;

    const int mbase = (wave >> 2) * 16;      // 2 M-tiles cover Q=32
    const int nbase = (wave & 3) * 16;       // 4 N-tiles cover 64 doc cols
    const int nt    = wave & 3;

    // Per-thread constants (fully unrolled -> VGPRs)
    const float alpha = alpha_p[0];
    float muv[NBINS], nis[NBINS];
#pragma unroll
    for (int k = 0; k < NBINS; ++k) {
        muv[k] = mu_p[k];
        float sg = sigma_p[k];
        nis[k] = -0.5f / (sg * sg);
    }
    // Inverse norms for this wave's 8 C-rows (fixed per lane-half)
    float qinv[8];
#pragma unroll
    for (int r = 0; r < 8; ++r)
        qinv[r] = invq[b * QLEN + mbase + r + 8 * half];

    // Zero accumulators
    for (int i = tid; i < 2 * 4 * QLEN * NBINS; i += 256) accN[i] = 0.f;

    // Stage raw query rows (async) + doc tile 0 (async), one fence
    for (int c = tid; c < QLEN * CHK_B; c += 256) {
        int r = c / CHK_B, q4 = (c - r * CHK_B) * 4;
        int tok = iq[b * QLEN + r];
        cp16_to_lds(emb + (size_t)tok * EDIM + q4, As + r * EP + q4);
    }
    for (int c = tid; c < DTILE * CHK_B; c += 256) {
        int r = c / CHK_B, q4 = (c - r * CHK_B) * 4;
        int tok = idd[b * DLEN + r];
        cp16_to_lds(emb + (size_t)tok * EDIM + q4, Bs + r * EP + q4);
    }
    async_fence_and_barrier();

    for (int t = 0; t < DLEN / DTILE; ++t) {
        const int d0 = t * DTILE;
        // Single LDS base + computed offsets keeps addrspace inference intact
        const int curOff = (t & 1) * (DTILE * EP);
        const int nxtOff = ((t + 1) & 1) * (DTILE * EP);

        // Prefetch next doc tile into the other buffer (overlaps with WMMA)
        if (t + 1 < DLEN / DTILE) {
            const int dn = d0 + DTILE;
            for (int c = tid; c < DTILE * CHK_B; c += 256) {
                int r = c / CHK_B, q4 = (c - r * CHK_B) * 4;
                int tok = idd[b * DLEN + dn + r];
                cp16_to_lds(emb + (size_t)tok * EDIM + q4,
                            Bs + nxtOff + r * EP + q4);
            }
        }

        // 16x16 f32 WMMA over K=300 (75 steps of K=4)
        const float* arow = &As[(mbase + l16) * EP];
        const float* brow = &Bs[curOff + (nbase + l16) * EP];
        v8f cc = {};
#pragma unroll 5
        for (int k0 = 0; k0 < EDIM; k0 += 4) {
            v2f a  = *(const v2f*)(arow + k0 + 2 * half);
            v2f bv = *(const v2f*)(brow + k0 + 2 * half);
            cc = __builtin_amdgcn_wmma_f32_16x16x4_f32(
                false, a, false, bv, (short)0, cc, false, false);
        }

        // Fused kernel pooling. Lane column = d0+nbase+l16, rows = mbase+r+8*half.
        const int dcol = d0 + nbase + l16;
        const float dinv = invd[b * DLEN + dcol];
        const float mdv  = mask_d[b * DLEN + dcol];
#pragma unroll
        for (int r = 0; r < 8; ++r) {
            float s = cc[r] * (qinv[r] * dinv);          // fold L2 norms here
            s = fminf(fmaxf(s, -1.0f + 1e-7f), 1.0f - 1e-7f);
            float tr = acosf(s) * 0.31830988618379067f;  // /pi
            int qrow = mbase + r + 8 * half;
#pragma unroll
            for (int k = 0; k < NBINS; ++k) {
                float df  = tr - muv[k];
                float pv  = __expf(df * df * nis[k]) * mdv;  // fast v_exp_f32
                float tpv = __expf(alpha * pv);
                float num = pv * tpv;
                float den = tpv;
#pragma unroll
                for (int off = 8; off >= 1; off >>= 1) {
                    num += __shfl_xor(num, off);
                    den += __shfl_xor(den, off);
                }
                if (l16 == 0) {
                    int idx = nt * (QLEN * NBINS) + qrow * NBINS + k;
                    accN[idx] += num;
                    accD[idx] += den;
                }
            }
        }

        // Next tile staged + all waves done reading -> safe to swap buffers
        async_fence_and_barrier();
    }

    // Final: pooling ratio -> log -> idf attention -> dense -> softsign
    if (tid < QLEN * NBINS) {
        float num = 0.f, den = 0.f;
#pragma unroll
        for (int t = 0; t < 4; ++t) {
            num += accN[t * (QLEN * NBINS) + tid];
            den += accD[t * (QLEN * NBINS) + tid];
        }
        float pool = fmaxf(num / den, 1e-10f);
        int q = tid / NBINS;
        int tok = iq[b * QLEN + q];
        float attn = mask_q[b * QLEN + q] * (attn_w[tok] * idf_w[0] + idf_b[0]);
        red[tid] = logf(pool) * attn * 0.01f;
    }
    __syncthreads();
    if (tid < NBINS) {
        float f = 0.f;
        for (int q = 0; q < QLEN; ++q) f += red[q * NBINS + tid];
        red[QLEN * NBINS + tid] = f * dense_w[tid];
    }
    __syncthreads();
    if (tid == 0) {
        float sc = dense_b[0];
#pragma unroll
        for (int k = 0; k < NBINS; ++k) sc += red[QLEN * NBINS + k];
        out[b] = sc / (1.0f + fabsf(sc));
    }
}

// ---------------------------------------------------------------------------
extern "C" void kernel_launch(void* const* d_in, const int* in_sizes, int n_in,
                              void* d_out, int out_size, void* d_ws, size_t ws_size,
                              hipStream_t stream)
{
    const int*   iq     = (const int*)  d_in[0];
    const int*   idd    = (const int*)  d_in[1];
    const float* mask_q = (const float*)d_in[2];
    const float* mask_d = (const float*)d_in[3];
    const float* emb    = (const float*)d_in[4];
    const float* attn_w = (const float*)d_in[5];
    const float* idf_w  = (const float*)d_in[6];
    const float* idf_b  = (const float*)d_in[7];
    const float* alpha  = (const float*)d_in[8];
    const float* mu     = (const float*)d_in[9];
    const float* sigma  = (const float*)d_in[10];
    const float* dw     = (const float*)d_in[11];
    const float* db     = (const float*)d_in[12];

    float* inv = (float*)d_ws;                 // [2048 q | 65536 d] inverse norms

    token_inv_norms<<<NTOK / 8, 256, 0, stream>>>(iq, idd, emb, inv);

    size_t smb = (size_t)(QLEN * EP + 2 * DTILE * EP + 2 * 4 * QLEN * NBINS + 384)
               * sizeof(float);               // ~207 KB, under 320 KB/WGP
    knrm_forward<<<BB, 256, smb, stream>>>(
        iq, idd, mask_q, mask_d, emb, attn_w, idf_w, idf_b,
        alpha, mu, sigma, dw, db, inv, inv + BB * QLEN, (float*)d_out);
}